// WindowAttention_7550552506833
// MI455X (gfx1250) — compile-verified
//
#include <hip/hip_runtime.h>

// ---------------------------------------------------------------------------
// Fused Swin window-attention for MI455X (gfx1250, wave32, WMMA f32 16x16x4).
// One block = one window (n=49 tokens, C=128). Everything stays in LDS:
//   phase1: qkv = X @ qkv_w + b         (WMMA, result -> LDS 64x384)
//   phase2: att = q k^T + bias + mask   (WMMA, -> LDS att buffer)
//   phase3: softmax rows                (VALU, bank-conflict-free rotation)
//   phase4: x = att @ v                 (WMMA)
//   phase5: out = x @ proj_w + proj_b   (WMMA, direct global store)
// ---------------------------------------------------------------------------

typedef float v2f __attribute__((ext_vector_type(2)));
typedef float v8f __attribute__((ext_vector_type(8)));

#define NTOK   49
#define NPAD   64
#define CDIM   128
#define NHEAD  4
#define HDIM   32
#define NW     64
#define QKV_N  384
#define ATT_HEAD (NPAD * NPAD)          // 4096 floats per head
#define LDS_BYTES (NPAD*QKV_N*4 + NHEAD*ATT_HEAD*4)   // 98304 + 65536 = 163840

__device__ __forceinline__ v8f wmma4(v2f a, v2f b, v8f c) {
  // D = A(16x4,f32) * B(4x16,f32) + C(16x16,f32)
  return __builtin_amdgcn_wmma_f32_16x16x4_f32(false, a, false, b,
                                               (short)0, c, false, false);
}

__global__ __launch_bounds__(256)
void fused_window_attn(const float* __restrict__ x_in,
                       const float* __restrict__ mask,
                       const float* __restrict__ qkv_w,
                       const float* __restrict__ qkv_b,
                       const float* __restrict__ proj_w,
                       const float* __restrict__ proj_b,
                       const float* __restrict__ rpbt,
                       float* __restrict__ out)
{
  extern __shared__ float smem[];
  float* QKVs = smem;                        // [64][384]  96 KB
  float* Xs   = smem + NPAD * QKV_N;         // [64][128]  (phase 0/1 only)
  float* attB = smem + NPAD * QKV_N;         // [4][64][64] (phase 2-4, overlays Xs)
  float* xS   = smem;                        // [64][128]  (phase 5, overlays QKVs)

  const int b    = blockIdx.x;
  const int tid  = threadIdx.x;
  const int wave = tid >> 5;
  const int lane = tid & 31;
  const int hw   = lane >> 4;                // half-wave (0/1)
  const int l16  = lane & 15;
  const int kofs = hw * 2;                   // A/B K-offset for this half-wave

  // ---- Phase 0: stage input tile, zero-pad rows 49..63; prefetch weights ---
  const float* xin = x_in + (size_t)b * (NTOK * CDIM);
  for (int idx = tid; idx < NPAD * CDIM; idx += 256) {
    int r = idx >> 7;
    Xs[idx] = (r < NTOK) ? xin[idx] : 0.0f;
  }
  for (int i = tid; i < (CDIM * QKV_N) / 32; i += 256)   // 128B granules
    __builtin_prefetch(qkv_w + i * 32, 0, 1);
  for (int i = tid; i < (CDIM * CDIM) / 32; i += 256)
    __builtin_prefetch(proj_w + i * 32, 0, 1);
  __syncthreads();

  // ---- Phase 1: QKV GEMM: [64x128] @ [128x384], wave owns 3 N-tiles -------
  const int nbase = wave * 48;
  v8f accQ[4][3];
  #pragma unroll
  for (int m = 0; m < 4; ++m)
    #pragma unroll
    for (int j = 0; j < 3; ++j)
      accQ[m][j] = (v8f){0,0,0,0,0,0,0,0};

  for (int kk = 0; kk < CDIM; kk += 4) {
    v2f a[4], bw[3];
    #pragma unroll
    for (int m = 0; m < 4; ++m)
      a[m] = *(const v2f*)&Xs[(m * 16 + l16) * CDIM + kk + kofs];
    #pragma unroll
    for (int j = 0; j < 3; ++j) {
      const float* wp = qkv_w + (size_t)(kk + kofs) * QKV_N + nbase + j * 16 + l16;
      bw[j].x = wp[0];
      bw[j].y = wp[QKV_N];
    }
    #pragma unroll
    for (int m = 0; m < 4; ++m)
      #pragma unroll
      for (int j = 0; j < 3; ++j)
        accQ[m][j] = wmma4(a[m], bw[j], accQ[m][j]);
  }

  // epilogue: +bias, scale q, park in LDS
  const float qscale = 0.17677669529663687f;   // 32^-0.5
  #pragma unroll
  for (int j = 0; j < 3; ++j) {
    int col = nbase + j * 16 + l16;
    float bias = qkv_b[col];
    bool isq = (col < CDIM);
    #pragma unroll
    for (int m = 0; m < 4; ++m)
      #pragma unroll
      for (int r = 0; r < 8; ++r) {
        int row = m * 16 + hw * 8 + r;
        float v = accQ[m][j][r] + bias;
        if (isq) v *= qscale;
        QKVs[row * QKV_N + col] = v;
      }
  }
  __syncthreads();

  // ---- Phase 2: scores = q k^T (+rel-pos bias, +mask). 2 waves per head ---
  const int h   = wave >> 1;
  const int sub = wave & 1;
  v8f accS[2][4];
  #pragma unroll
  for (int mi = 0; mi < 2; ++mi)
    #pragma unroll
    for (int ni = 0; ni < 4; ++ni)
      accS[mi][ni] = (v8f){0,0,0,0,0,0,0,0};

  for (int kk = 0; kk < HDIM; kk += 4) {
    v2f a[2], bk[4];
    #pragma unroll
    for (int mi = 0; mi < 2; ++mi)
      a[mi] = *(const v2f*)&QKVs[((sub * 2 + mi) * 16 + l16) * QKV_N
                                 + h * HDIM + kk + kofs];
    #pragma unroll
    for (int ni = 0; ni < 4; ++ni)
      bk[ni] = *(const v2f*)&QKVs[(ni * 16 + l16) * QKV_N
                                  + CDIM + h * HDIM + kk + kofs];
    #pragma unroll
    for (int mi = 0; mi < 2; ++mi)
      #pragma unroll
      for (int ni = 0; ni < 4; ++ni)
        accS[mi][ni] = wmma4(a[mi], bk[ni], accS[mi][ni]);
  }

  const float* mrow = mask + (size_t)(b & (NW - 1)) * (NTOK * NTOK);
  #pragma unroll
  for (int mi = 0; mi < 2; ++mi)
    #pragma unroll
    for (int ni = 0; ni < 4; ++ni)
      #pragma unroll
      for (int r = 0; r < 8; ++r) {
        int i = (sub * 2 + mi) * 16 + hw * 8 + r;
        int j = ni * 16 + l16;
        float v;
        if (i < NTOK && j < NTOK) {
          int ridx = 13 * (i / 7 - j / 7 + 6) + (i % 7 - j % 7 + 6);
          v = accS[mi][ni][r] + rpbt[ridx * NHEAD + h] + mrow[i * NTOK + j];
        } else {
          v = -1e30f;                         // padding -> softmax weight 0
        }
        attB[h * ATT_HEAD + i * NPAD + j] = v;
      }
  __syncthreads();

  // ---- Phase 3: softmax, one row per thread (4 heads x 64 rows = 256) -----
  {
    int hh  = tid >> 6;
    int i   = tid & 63;
    int rot = i & 63;                         // rotate -> conflict-free banks
    float* row = attB + hh * ATT_HEAD + i * NPAD;
    float mx = -3e30f;
    for (int j = 0; j < NPAD; ++j) mx = fmaxf(mx, row[(j + rot) & 63]);
    float s = 0.0f;
    for (int j = 0; j < NPAD; ++j) {
      int c = (j + rot) & 63;
      float e = __expf(row[c] - mx);
      row[c] = e;
      s += e;
    }
    float inv = 1.0f / s;
    for (int j = 0; j < NPAD; ++j) { int c = (j + rot) & 63; row[c] *= inv; }
  }
  __syncthreads();

  // ---- Phase 4: x = att @ v  (K runs over 64; padded cols are exactly 0) --
  v8f accX[2][2];
  #pragma unroll
  for (int mi = 0; mi < 2; ++mi)
    #pragma unroll
    for (int ni = 0; ni < 2; ++ni)
      accX[mi][ni] = (v8f){0,0,0,0,0,0,0,0};

  for (int kk = 0; kk < NPAD; kk += 4) {
    v2f a[2], bv[2];
    #pragma unroll
    for (int mi = 0; mi < 2; ++mi)
      a[mi] = *(const v2f*)&attB[h * ATT_HEAD
                                 + ((sub * 2 + mi) * 16 + l16) * NPAD + kk + kofs];
    #pragma unroll
    for (int ni = 0; ni < 2; ++ni) {
      int krow = kk + kofs;
      const float* vp = QKVs + krow * QKV_N + 2 * CDIM + h * HDIM + ni * 16 + l16;
      bv[ni].x = vp[0];
      bv[ni].y = vp[QKV_N];
    }
    #pragma unroll
    for (int mi = 0; mi < 2; ++mi)
      #pragma unroll
      for (int ni = 0; ni < 2; ++ni)
        accX[mi][ni] = wmma4(a[mi], bv[ni], accX[mi][ni]);
  }
  __syncthreads();                            // all waves done reading QKVs

  #pragma unroll
  for (int mi = 0; mi < 2; ++mi)
    #pragma unroll
    for (int ni = 0; ni < 2; ++ni)
      #pragma unroll
      for (int r = 0; r < 8; ++r) {
        int i  = (sub * 2 + mi) * 16 + hw * 8 + r;
        int cc = h * HDIM + ni * 16 + l16;
        xS[i * CDIM + cc] = accX[mi][ni][r];  // overlays QKVs
      }
  __syncthreads();

  // ---- Phase 5: out = x @ proj_w + proj_b. Wave owns one 16-col N-tile ----
  const int n0 = wave * 16;
  v8f accP[4];
  #pragma unroll
  for (int m = 0; m < 4; ++m) accP[m] = (v8f){0,0,0,0,0,0,0,0};

  for (int kk = 0; kk < CDIM; kk += 4) {
    v2f a[4], bp;
    #pragma unroll
    for (int m = 0; m < 4; ++m)
      a[m] = *(const v2f*)&xS[(m * 16 + l16) * CDIM + kk + kofs];
    const float* wp = proj_w + (kk + kofs) * CDIM + n0 + l16;
    bp.x = wp[0];
    bp.y = wp[CDIM];
    #pragma unroll
    for (int m = 0; m < 4; ++m)
      accP[m] = wmma4(a[m], bp, accP[m]);
  }

  float pb = proj_b[n0 + l16];
  float* outp = out + (size_t)b * (NTOK * CDIM);
  #pragma unroll
  for (int m = 0; m < 4; ++m)
    #pragma unroll
    for (int r = 0; r < 8; ++r) {
      int i = m * 16 + hw * 8 + r;
      if (i < NTOK) outp[i * CDIM + n0 + l16] = accP[m][r] + pb;
    }
}

extern "C" void kernel_launch(void* const* d_in, const int* in_sizes, int n_in,
                              void* d_out, int out_size, void* d_ws, size_t ws_size,
                              hipStream_t stream) {
  (void)in_sizes; (void)n_in; (void)out_size; (void)d_ws; (void)ws_size;
  const float* x_in   = (const float*)d_in[0];
  const float* mask   = (const float*)d_in[1];
  const float* qkv_w  = (const float*)d_in[2];
  const float* qkv_b  = (const float*)d_in[3];
  const float* proj_w = (const float*)d_in[4];
  const float* proj_b = (const float*)d_in[5];
  const float* rpbt   = (const float*)d_in[6];
  float* out = (float*)d_out;

  fused_window_attn<<<dim3(4096), dim3(256), LDS_BYTES, stream>>>(
      x_in, mask, qkv_w, qkv_b, proj_w, proj_b, rpbt, out);
}